// Quantizer_6923487281290
// MI455X (gfx1250) — compile-verified
//
#include <hip/hip_runtime.h>
#include <stdint.h>

#define NUM_E  1024
#define DIM    64
#define HW     4096                      // 64*64
#define NROWS  65536                     // 16*4096 flattened z vectors
#define TOTAL  4194304                   // 16*64*4096 elements of z / z_q

// d_out layout (floats, reference return order):
//   [0]                      loss
//   [1 .. 1+TOTAL)           z_q_out (NCHW)
//   [ME_OFF .. +NROWS*NUM_E) min_encodings (one-hot)
//   [IDX_OFF .. +NROWS)      train_indices (stored as float)
#define ZQ_OFF   ((size_t)1)
#define ME_OFF   ((size_t)(1 + TOTAL))                    // 4194305
#define IDX_OFF  (ME_OFF + (size_t)NROWS * NUM_E)         // 71303169

typedef __attribute__((ext_vector_type(2))) float v2f;
typedef __attribute__((ext_vector_type(8))) float v8f;

// ---------------------------------------------------------------- kernel 0
__global__ void __launch_bounds__(256)
k_zero(float* __restrict__ out) {
    const size_t stride = (size_t)gridDim.x * blockDim.x;
    size_t i = (size_t)blockIdx.x * blockDim.x + threadIdx.x;
    const size_t me_elems = (size_t)NROWS * NUM_E;        // 67108864
    const size_t n4 = (me_elems - 3) / 4;                 // aligned float4 body
    if (i == 0) {
        out[0] = 0.0f;                                    // loss accumulator
        out[ME_OFF + 0] = 0.0f;                           // unaligned head
        out[ME_OFF + 1] = 0.0f;
        out[ME_OFF + 2] = 0.0f;
        out[ME_OFF + 3 + n4 * 4] = 0.0f;                  // tail
    }
    float4* me = (float4*)(out + ME_OFF + 3);             // 16B-aligned
    for (size_t j = i; j < n4; j += stride)
        me[j] = make_float4(0.f, 0.f, 0.f, 0.f);
}

// ---------------------------------------------------------------- kernel 1
__global__ void __launch_bounds__(256)
k_ekk(const float* __restrict__ emb, float* __restrict__ ekk) {
    int k = blockIdx.x * blockDim.x + threadIdx.x;
    if (k < NUM_E) {
        float s = 0.f;
        #pragma unroll
        for (int c = 0; c < DIM; ++c) {
            float v = emb[k * DIM + c];
            s += v * v;
        }
        ekk[k] = s;
    }
}

// ---------------------------------------------------------------- kernel 2
// One wave per 16-row tile of z. Computes dot(z_row, e_code) for all 1024
// codes via V_WMMA_F32_16X16X4_F32 and tracks the running argmin of
// (||e||^2 - 2*dot) per row.
__global__ void __launch_bounds__(256)
k_argmin(const float* __restrict__ z, const float* __restrict__ emb,
         const float* __restrict__ ekk, int* __restrict__ idx_ws,
         float* __restrict__ out) {
    const int lane = threadIdx.x & 31;
    const int wave = blockIdx.x * (blockDim.x >> 5) + (threadIdx.x >> 5);
    const int m0   = wave * 16;                 // first flattened row of tile
    const int b    = m0 / HW;                   // batch (tiles never straddle)
    const int hw0  = m0 % HW;
    const int sub  = lane & 15;                 // 0..15
    const int hi   = lane >> 4;                 // 0 or 1

    // A operand (16x4 f32 WMMA layout): lanes 0..15 = rows m0+sub, K={4s,4s+1};
    // lanes 16..31 = same rows, K={4s+2,4s+3}. z is NCHW: channel stride = HW.
    const float* zrow = z + (size_t)b * DIM * HW + (size_t)(hw0 + sub);
    v2f a[16];
    #pragma unroll
    for (int s = 0; s < 16; ++s) {
        int c = s * 4 + hi * 2;
        a[s].x = zrow[(size_t)c * HW];
        a[s].y = zrow[(size_t)(c + 1) * HW];
    }

    float minVal[8];
    int   minIdx[8];
    #pragma unroll
    for (int r = 0; r < 8; ++r) { minVal[r] = 3.4e38f; minIdx[r] = 0; }

    for (int chunk = 0; chunk < 64; ++chunk) {
        const int codeN = chunk * 16 + sub;     // this lane's column code
        // B operand (4x16 layout): lanes 0..15 = col sub K={4s,4s+1},
        // lanes 16..31 = col sub K={4s+2,4s+3}. Embedding is row-major [1024,64].
        const float* erow = emb + (size_t)codeN * DIM + hi * 2;
        v2f bm[16];
        #pragma unroll
        for (int s = 0; s < 16; ++s)
            bm[s] = *(const v2f*)(erow + s * 4);

        v8f acc = {};
        #pragma unroll
        for (int s = 0; s < 16; ++s)
            acc = __builtin_amdgcn_wmma_f32_16x16x4_f32(
                false, a[s], false, bm[s], (short)0, acc, false, false);

        const float ek = ekk[codeN];
        #pragma unroll
        for (int r = 0; r < 8; ++r) {
            float d = ek - 2.0f * acc[r];       // argmin-equivalent distance
            if (d < minVal[r]) { minVal[r] = d; minIdx[r] = codeN; }
        }
    }

    // Butterfly reduce across the 16 column-lanes of each half-wave.
    // Tie-break on smaller index to match jnp.argmin (first occurrence).
    #pragma unroll
    for (int m = 1; m < 16; m <<= 1) {
        #pragma unroll
        for (int r = 0; r < 8; ++r) {
            float ov = __shfl_xor(minVal[r], m, 32);
            int   oi = __shfl_xor(minIdx[r], m, 32);
            if (ov < minVal[r] || (ov == minVal[r] && oi < minIdx[r])) {
                minVal[r] = ov; minIdx[r] = oi;
            }
        }
    }

    if (sub == 0) {                             // lanes 0 and 16 write results
        #pragma unroll
        for (int r = 0; r < 8; ++r) {
            int n = m0 + hi * 8 + r;
            int k = minIdx[r];
            idx_ws[n] = k;
            out[IDX_OFF + (size_t)n] = (float)k;
            out[ME_OFF + (size_t)n * NUM_E + (size_t)k] = 1.0f;
        }
    }
}

// ---------------------------------------------------------------- kernel 3
// Gather codebook rows into NCHW output (coalesced over hw) + loss reduction.
__global__ void __launch_bounds__(256)
k_gather(const float* __restrict__ z, const float* __restrict__ emb,
         const int* __restrict__ idx_ws, float* __restrict__ out) {
    __shared__ float red[256];
    size_t t  = (size_t)blockIdx.x * blockDim.x + threadIdx.x;  // NCHW flat
    size_t hw = t & (size_t)(HW - 1);
    size_t bc = t >> 12;                     // b*DIM + c
    size_t c  = bc & (size_t)(DIM - 1);
    size_t b  = bc >> 6;
    int    k  = idx_ws[b * HW + hw];
    float  zq = emb[(size_t)k * DIM + c];
    float  zv = z[t];
    out[ZQ_OFF + t] = zq;
    float d = zq - zv;
    red[threadIdx.x] = d * d;
    __syncthreads();
    for (int s = 128; s > 0; s >>= 1) {
        if (threadIdx.x < (unsigned)s) red[threadIdx.x] += red[threadIdx.x + s];
        __syncthreads();
    }
    if (threadIdx.x == 0)
        atomicAdd(out, red[0] * (1.25f / (float)TOTAL));  // (1+beta)*mean
}

// ---------------------------------------------------------------- launcher
extern "C" void kernel_launch(void* const* d_in, const int* in_sizes, int n_in,
                              void* d_out, int out_size, void* d_ws, size_t ws_size,
                              hipStream_t stream) {
    (void)in_sizes; (void)n_in; (void)out_size; (void)ws_size;
    const float* z   = (const float*)d_in[0];
    const float* emb = (const float*)d_in[1];
    float* out       = (float*)d_out;
    int*   idx_ws    = (int*)d_ws;                 // 65536 ints
    float* ekk       = (float*)d_ws + NROWS;       // 1024 floats after indices

    k_zero  <<<4096, 256, 0, stream>>>(out);
    k_ekk   <<<NUM_E / 256, 256, 0, stream>>>(emb, ekk);
    k_argmin<<<NROWS / 16 / 8, 256, 0, stream>>>(z, emb, ekk, idx_ws, out);
    k_gather<<<TOTAL / 256, 256, 0, stream>>>(z, emb, idx_ws, out);
}